// GeneralModelforMaskedLM_35811437314826
// MI455X (gfx1250) — compile-verified
//
#include <hip/hip_runtime.h>
#include <math.h>
#include <stdint.h>

// Problem shape (fixed by the reference).
#define BN   4
#define SN   256
#define VN   50257
#define NTHR 256          // 8 wave32 per block
#define TILE 1536         // floats per TDM tile (6 KB); 2 bufs x 2 arrays = 24 KB LDS
#define NTILE ((VN + TILE - 1) / TILE)   // 33 tiles (last is partial, TDM zero-fills)
#define EPSF 1e-20f

typedef unsigned int u32;
typedef u32   u32x4 __attribute__((ext_vector_type(4)));
typedef int   i32x4 __attribute__((ext_vector_type(4)));
typedef int   i32x8 __attribute__((ext_vector_type(8)));
typedef float f32x4 __attribute__((ext_vector_type(4)));

// Issue one TDM 1-D row-tile load: copies a row tile from global memory at gptr to LDS at lds_byte_addr.
// Descriptor packing per CDNA5 ISA ch.8:
//   group0: count=1 | lds_addr | global_addr[56:0] | type=2
//   group1: data_size=4B, tensor_dim0=n_valid (OOB reads -> 0), tensor_dim1=1,
//           tile_dim0=TILE, dim0_stride=n_valid, everything else 0.
__device__ __forceinline__ void tdm_load_row_tile(u32 lds_byte_addr,
                                                  const float* gptr,
                                                  u32 n_valid) {
  unsigned long long ga = (unsigned long long)(uintptr_t)gptr;
  u32x4 g0;
  g0.x = 1u;                                              // count = 1
  g0.y = lds_byte_addr;                                   // LDS dest (bytes)
  g0.z = (u32)ga;                                         // global_addr[31:0]
  g0.w = ((u32)(ga >> 32) & 0x01FFFFFFu) | (2u << 30);    // ga[56:32] | type=2
  i32x8 g1;
  g1[0] = (int)(2u << 16);                                // data_size = 4 bytes
  g1[1] = (int)((n_valid & 0xFFFFu) << 16);               // tensor_dim0[15:0]
  g1[2] = (int)((n_valid >> 16) & 0xFFFFu) | (1 << 16);   // tensor_dim0[31:16], tensor_dim1=1
  g1[3] = (int)((u32)TILE << 16);                         // tile_dim0 = TILE
  g1[4] = 0;                                              // tile_dim1 = tile_dim2 = 0 (unused)
  g1[5] = (int)n_valid;                                   // tensor_dim0_stride[31:0]
  g1[6] = 0;
  g1[7] = 0;
  i32x4 z4 = (i32x4)0;
#if __has_include(<hip/amd_detail/amd_gfx1250_TDM.h>)
  i32x8 z8 = (i32x8)0;
  __builtin_amdgcn_tensor_load_to_lds(g0, g1, z4, z4, z8, 0);
#else
  __builtin_amdgcn_tensor_load_to_lds(g0, g1, z4, z4, 0);
#endif
}

// Write a one-hot row (zeros except orow[hot] = val) with NT b128 stores.
// Row base is only guaranteed 4B-aligned (V odd), so: scalar prologue to 16B
// alignment, float4 NT body, scalar tail.
__device__ __forceinline__ void write_onehot_row(float* __restrict__ orow,
                                                 int hot, float val, int tid) {
  const int pre  = ((int)(16u - ((u32)(uintptr_t)orow & 15u)) & 15) >> 2;  // 0..3
  const int nvec = (VN - pre) >> 2;
  const int vend = pre + (nvec << 2);

  for (int v = tid; v < pre; v += NTHR)
    __builtin_nontemporal_store((v == hot) ? val : 0.0f, orow + v);

  f32x4* __restrict__ ov = (f32x4*)(orow + pre);
  for (int q = tid; q < nvec; q += NTHR) {
    const int v = pre + (q << 2);
    f32x4 w;
    w.x = (v + 0 == hot) ? val : 0.0f;
    w.y = (v + 1 == hot) ? val : 0.0f;
    w.z = (v + 2 == hot) ? val : 0.0f;
    w.w = (v + 3 == hot) ? val : 0.0f;
    __builtin_nontemporal_store(w, ov + q);
  }

  for (int v = vend + tid; v < VN; v += NTHR)
    __builtin_nontemporal_store((v == hot) ? val : 0.0f, orow + v);
}

__global__ __launch_bounds__(NTHR)
void mlm_gumbel_onehot_kernel(const float* __restrict__ scores,
                              const float* __restrict__ gumbel,
                              const int* __restrict__ ids,
                              const int* __restrict__ mask,
                              float* __restrict__ out) {
  __shared__ float lds_s[2][TILE];   // staged prediction_scores tile
  __shared__ float lds_g[2][TILE];   // staged gumbel_u tile
  __shared__ float r_m[NTHR];
  __shared__ float r_s[NTHR];
  __shared__ int   r_a[NTHR];

  const int tid = threadIdx.x;
  const int row = blockIdx.x;        // b*SN + t
  const int b   = row >> 8;          // SN == 256
  const int t   = row & 255;

  float* orow = out + (size_t)row * VN;
  const int mval = mask[row];        // uniform within block

  if (mval == 0) {
    // Unmasked position: exact one-hot of input_ids.
    write_onehot_row(orow, ids[row], 1.0f, tid);
    return;
  }

  // ---- exclusive prefix count of masked positions before t (compact index) ----
  r_a[tid] = (tid < t) ? mask[(b << 8) + tid] : 0;
  __syncthreads();
  for (int off = NTHR >> 1; off > 0; off >>= 1) {
    if (tid < off) r_a[tid] += r_a[tid + off];
    __syncthreads();
  }
  const int cidx = __builtin_amdgcn_readfirstlane(r_a[0]);  // keep scalar for TDM addrs
  __syncthreads();  // r_a reused below

  const float* srow = scores + ((size_t)(b << 8) + (size_t)t) * VN;
  const float* grow = gumbel + ((size_t)(b << 8) + (size_t)cidx) * VN;

  const bool issuer = (tid < 32);    // wave 0 issues TDM ops (uniform branch)

  if (issuer) {
    tdm_load_row_tile((u32)(uintptr_t)&lds_s[0][0], srow, (u32)VN);
    tdm_load_row_tile((u32)(uintptr_t)&lds_g[0][0], grow, (u32)VN);
  }

  // ---- online softmax (max, argmax, sum of exp) over V, TDM double-buffered ----
  float m   = -INFINITY;
  float sum = 0.0f;
  int   am  = 0;

  for (int k = 0; k < NTILE; ++k) {
    const int buf  = k & 1;
    const int base = k * TILE;
    const int cnt  = (VN - base < TILE) ? (VN - base) : TILE;

    if (issuer) {
      if (k + 1 < NTILE) {
        const int nb    = (k + 1) & 1;
        const int nbase = (k + 1) * TILE;
        tdm_load_row_tile((u32)(uintptr_t)&lds_s[nb][0], srow + nbase, (u32)(VN - nbase));
        tdm_load_row_tile((u32)(uintptr_t)&lds_g[nb][0], grow + nbase, (u32)(VN - nbase));
        __builtin_amdgcn_s_wait_tensorcnt(2);  // tile k done; k+1 (2 ops) in flight
      } else {
        __builtin_amdgcn_s_wait_tensorcnt(0);
      }
    }
    __syncthreads();                           // TDM data visible to all 8 waves

    if (cnt == TILE) {
      // Full tile: 16B/lane LDS reads -> ds_load_b128, conflict-free.
      const f32x4* ps = (const f32x4*)&lds_s[buf][0];
      const f32x4* pg = (const f32x4*)&lds_g[buf][0];
      for (int q = tid; q < TILE / 4; q += NTHR) {
        const f32x4 s4 = ps[q];
        const f32x4 u4 = pg[q];
        const int j0 = base + (q << 2);
#pragma unroll
        for (int e = 0; e < 4; ++e) {
          const float sc = (e == 0) ? s4.x : (e == 1) ? s4.y : (e == 2) ? s4.z : s4.w;
          const float u  = (e == 0) ? u4.x : (e == 1) ? u4.y : (e == 2) ? u4.z : u4.w;
          const float gn = -logf(-logf(u + EPSF) + EPSF);
          const float z  = (sc + gn) * 2.0f;   // divide by TEMPERATURE = 0.5
          if (z > m) {                         // strict '>' => first occurrence
            sum = sum * expf(m - z) + 1.0f;
            am  = j0 + e;
            m   = z;
          } else {
            sum += expf(z - m);
          }
        }
      }
    } else {
      // Tail tile: scalar.
      for (int j = tid; j < cnt; j += NTHR) {
        const float sc = lds_s[buf][j];
        const float u  = lds_g[buf][j];
        const float gn = -logf(-logf(u + EPSF) + EPSF);
        const float z  = (sc + gn) * 2.0f;
        if (z > m) {
          sum = sum * expf(m - z) + 1.0f;
          am  = base + j;
          m   = z;
        } else {
          sum += expf(z - m);
        }
      }
    }
    __syncthreads();                           // done reading buf before refill
  }

  // ---- block reduction of (m, sum, argmax); ties -> smaller index (jnp.argmax) ----
  r_m[tid] = m; r_s[tid] = sum; r_a[tid] = am;
  __syncthreads();
  for (int off = NTHR >> 1; off > 0; off >>= 1) {
    if (tid < off) {
      float m1 = r_m[tid],       s1 = r_s[tid];       int a1 = r_a[tid];
      float m2 = r_m[tid + off], s2 = r_s[tid + off]; int a2 = r_a[tid + off];
      float mn, sn; int an;
      if (m1 > m2)      { mn = m1; sn = s1 + s2 * expf(m2 - m1); an = a1; }
      else if (m2 > m1) { mn = m2; sn = s2 + s1 * expf(m1 - m2); an = a2; }
      else              { mn = m1; sn = s1 + s2; an = (a1 < a2) ? a1 : a2; }
      r_m[tid] = mn; r_s[tid] = sn; r_a[tid] = an;
    }
    __syncthreads();
  }

  const float y    = 1.0f / r_s[0];            // softmax value at the argmax
  const int   amax = r_a[0];
  const float val  = (1.0f - y) + y;           // straight-through: matches ref arithmetic
  write_onehot_row(orow, amax, val, tid);
}

extern "C" void kernel_launch(void* const* d_in, const int* in_sizes, int n_in,
                              void* d_out, int out_size, void* d_ws, size_t ws_size,
                              hipStream_t stream) {
  (void)in_sizes; (void)n_in; (void)out_size; (void)d_ws; (void)ws_size;
  const float* scores = (const float*)d_in[0];   // prediction_scores [B,S,V] f32
  const float* gumbel = (const float*)d_in[1];   // gumbel_u          [B,S,V] f32
  const int*   ids    = (const int*)d_in[2];     // input_ids         [B,S]   i32
  const int*   msk    = (const int*)d_in[3];     // my_attention_mask [B,S]   i32
  float* out = (float*)d_out;                    // [B*S, V] f32

  mlm_gumbel_onehot_kernel<<<dim3(BN * SN), dim3(NTHR), 0, stream>>>(
      scores, gumbel, ids, msk, out);
}